// MultiHeadAttention_1211180778282
// MI455X (gfx1250) — compile-verified
//
#include <hip/hip_runtime.h>
#include <hip/hip_bf16.h>
#include <math.h>

typedef __attribute__((ext_vector_type(2))) float v2f;
typedef __attribute__((ext_vector_type(8))) float v8f;
typedef __attribute__((ext_vector_type(4))) int   v4i_t;

#define DMODEL 768
#define NHEAD  12
#define HDIM   64
#define BATCH  2
#define SEQ    4096
#define MTOK   (BATCH * SEQ)      // 8192 rows
#define NEGBIG (-1e9f)

// gfx1250 async global->LDS copy (ASYNCcnt-tracked, no VGPR staging).
#define ASYNC_B128(gsrc, ldst)                                              \
  __builtin_amdgcn_global_load_async_to_lds_b128(                           \
      (v4i_t __attribute__((address_space(1)))*)(gsrc),                     \
      (v4i_t __attribute__((address_space(3)))*)(ldst), 0, 0)

// LDS row stride for 16-row fragment reads: bank = 4*(9*row mod 16) + c,
// distinct for all 16 rows -> conflict-free transposed access; keeps 16B align.
#define LSTR 36
#define KC   32                   // K-chunk depth staged per double-buffer step

// ---------------------------------------------------------------------------
// GEMM: Y = X @ W^T + bias.  X:[M,768], W:[768,768] row-major (torch Linear),
// Y[m][n] = sum_k X[m][k]*W[n][k] + bias[n].
// Block = 4 waves = 64x64 output tile. X/W K-chunks are double-buffered in
// LDS via the CDNA5 async global->LDS path; the 4 waves share the W tile.
// Each wave computes 16x64 with V_WMMA_F32_16X16X4_F32 fed from ds_load_b64.
// headLayout=1: store into [B, H, S, 64]; else plain [M, 768].
// ---------------------------------------------------------------------------
__global__ __launch_bounds__(128) void gemm_wmma(
    const float* __restrict__ X, const float* __restrict__ W,
    const float* __restrict__ bias, float* __restrict__ Y, int headLayout) {
  __shared__ float sX[2][64 * LSTR];
  __shared__ float sW[2][64 * LSTR];

  const int tid  = threadIdx.x;
  const int lane = tid & 31;
  const int wave = tid >> 5;
  const int col  = lane & 15;     // A: row m; B/C/D: column n
  const int half = lane >> 4;     // K/M sub-select per ISA layout
  const int mblk = blockIdx.x * 64;
  const int m0   = mblk + wave * 16;
  const int n0   = blockIdx.y * 64;

  // Cooperative copy mapping: 128 threads x (4+4) b128 ops per chunk.
  const int cr  = tid >> 1;            // 0..63 tile row
  const int cc  = (tid & 1) * 16;      // 0 or 16: half of the 32-float row

  auto issue_chunk = [&](int k0, int buf) {
    const float* gX = X + (size_t)(mblk + cr) * DMODEL + k0 + cc;
    const float* gW = W + (size_t)(n0 + cr) * DMODEL + k0 + cc;
    float* dX = &sX[buf][cr * LSTR + cc];
    float* dW = &sW[buf][cr * LSTR + cc];
#pragma unroll
    for (int i = 0; i < 4; ++i) ASYNC_B128(gX + 4 * i, dX + 4 * i);
#pragma unroll
    for (int i = 0; i < 4; ++i) ASYNC_B128(gW + 4 * i, dW + 4 * i);
  };

  v8f acc[4] = {};
  const int NCH = DMODEL / KC;    // 24 chunks
  issue_chunk(0, 0);

  for (int c = 0; c < NCH; ++c) {
    const int cur = c & 1;
    if (c + 1 < NCH) {
      issue_chunk((c + 1) * KC, cur ^ 1);
      __builtin_amdgcn_s_wait_asynccnt(8);  // chunk c landed, c+1 in flight
    } else {
      __builtin_amdgcn_s_wait_asynccnt(0);
    }
    __syncthreads();

#pragma unroll
    for (int kk = 0; kk < KC; kk += 4) {
      v2f a = *(const v2f*)(&sX[cur][(wave * 16 + col) * LSTR + kk + 2 * half]);
      v2f bf[4];
#pragma unroll
      for (int t = 0; t < 4; ++t)
        bf[t] = *(const v2f*)(&sW[cur][(16 * t + col) * LSTR + kk + 2 * half]);
#pragma unroll
      for (int t = 0; t < 4; ++t)
        acc[t] = __builtin_amdgcn_wmma_f32_16x16x4_f32(
            false, a, false, bf[t], (short)0, acc[t], false, false);
    }
    __syncthreads();   // all waves done with buf 'cur' before it is refilled
  }

#pragma unroll
  for (int t = 0; t < 4; ++t) {
    const int n = n0 + 16 * t + col;
    const float bv = bias[n];
#pragma unroll
    for (int r = 0; r < 8; ++r) {
      const int row = m0 + r + 8 * half;
      const float val = acc[t][r] + bv;
      if (headLayout) {
        const int b = row >> 12;          // row / 4096
        const int s = row & (SEQ - 1);
        const int h = n >> 6;             // n / 64
        const int dd = n & (HDIM - 1);
        Y[((((size_t)b * NHEAD + h) * SEQ + s) << 6) + dd] = val;
      } else {
        Y[(size_t)row * DMODEL + n] = val;
      }
    }
  }
}

// ---------------------------------------------------------------------------
// Flash attention. qh/kh/vh: [B,H,S,64]. Block = 4 waves = 64 queries of one
// (b,h). K/V chunks are double-buffered in LDS and filled with the CDNA5
// async global->LDS path (ASYNCcnt): chunk kb+1 streams in while chunk kb is
// consumed by the WMMAs. S^T = K @ Q^T puts one query per lane so the online
// softmax reduction is 7 VALU ops + one shfl_xor(16); O^T += V^T @ P^T routes
// P through 8 cross-half shuffles per chunk. Output -> attn:[B,S,768].
// ---------------------------------------------------------------------------
__global__ __launch_bounds__(128) void flash_attn(
    const float* __restrict__ qh, const float* __restrict__ kh,
    const float* __restrict__ vh, const int* __restrict__ pad_mask,
    float* __restrict__ attn) {
  __shared__ float sK[2][16 * 68];   // stride 68: conflict-free transposed reads
  __shared__ float sV[2][16 * 68];
  __shared__ float sM[2][16];

  const int tid  = threadIdx.x;
  const int wave = tid >> 5;
  const int lane = tid & 31;
  const int col  = lane & 15;     // query column owned by this lane
  const int half = lane >> 4;

  const int bh = blockIdx.y;
  const int b  = bh / NHEAD;
  const int h  = bh % NHEAD;
  const size_t base = (size_t)bh * SEQ * HDIM;
  const int q0 = blockIdx.x * 64 + wave * 16;

  // Per-thread source/dest for the cooperative chunk copy.
  const int cr  = tid >> 3;          // 0..15 key row
  const int cc8 = (tid & 7) * 8;     // 8 floats (2x b128) per thread per matrix

  auto issue_chunk = [&](int kb, int buf) {
    const float* gK = kh + base + (size_t)(kb * 16 + cr) * HDIM + cc8;
    const float* gV = vh + base + (size_t)(kb * 16 + cr) * HDIM + cc8;
    float* dK = &sK[buf][cr * 68 + cc8];
    float* dV = &sV[buf][cr * 68 + cc8];
    ASYNC_B128(gK, dK);
    ASYNC_B128(gK + 4, dK + 4);
    ASYNC_B128(gV, dV);
    ASYNC_B128(gV + 4, dV + 4);
    if (tid < 16)
      sM[buf][tid] = (pad_mask[b * SEQ + kb * 16 + tid] == 0) ? 1.0f : 0.0f;
  };

  // Q fragments for all 16 K-steps of the 64-deep dot product, scale folded in.
  v2f qf[16];
#pragma unroll
  for (int k0 = 0; k0 < 16; ++k0) {
    v2f t = *(const v2f*)(qh + base + (size_t)(q0 + col) * HDIM + k0 * 4 + 2 * half);
    qf[k0] = t * 0.125f;          // 1/sqrt(64)
  }

  float m_run = -INFINITY;
  float l_run = 0.0f;
  v8f acc[4] = {};                // O^T: d-rows (16t + r + 8*half) x query col

  const int NCH = SEQ / 16;       // 256 chunks
  issue_chunk(0, 0);

  for (int kb = 0; kb < NCH; ++kb) {
    const int cur = kb & 1;
    if (kb + 1 < NCH) {
      issue_chunk(kb + 1, cur ^ 1);
      __builtin_amdgcn_s_wait_asynccnt(4);  // chunk kb landed, kb+1 in flight
    } else {
      __builtin_amdgcn_s_wait_asynccnt(0);
    }
    __syncthreads();

    // S^T tile: A = K chunk (16 keys x 64d), B = Q^T -> lane col = query.
    v8f st = {};
#pragma unroll
    for (int k0 = 0; k0 < 16; ++k0) {
      v2f a = *(const v2f*)(&sK[cur][col * 68 + k0 * 4 + 2 * half]);
      st = __builtin_amdgcn_wmma_f32_16x16x4_f32(
          false, a, false, qf[k0], (short)0, st, false, false);
    }

    // Mask + online softmax. Lane holds 8 keys (rows r+8*half) of its query.
    float p[8];
    float mc = -INFINITY;
#pragma unroll
    for (int r = 0; r < 8; ++r) {
      float v = st[r];
      v = (sM[cur][r + 8 * half] > 0.5f) ? NEGBIG : v;
      p[r] = v;
      mc = fmaxf(mc, v);
    }
    mc = fmaxf(mc, __shfl_xor(mc, 16, 32));
    const float m_new = fmaxf(m_run, mc);
    const float alpha = __expf(m_run - m_new);
    float lc = 0.0f;
#pragma unroll
    for (int r = 0; r < 8; ++r) {
      p[r] = __expf(p[r] - m_new);
      lc += p[r];
    }
    lc += __shfl_xor(lc, 16, 32);
    l_run = l_run * alpha + lc;
    m_run = m_new;
#pragma unroll
    for (int t = 0; t < 4; ++t)
#pragma unroll
      for (int r = 0; r < 8; ++r) acc[t][r] *= alpha;

    // Route P^T rows so every lane can build its B fragment (cross-half shfl).
    float brow[16];
#pragma unroll
    for (int r = 0; r < 8; ++r) {
      const float other = __shfl_xor(p[r], 16, 32);
      brow[r]     = half ? other : p[r];   // global key row r   lives in half 0
      brow[r + 8] = half ? p[r] : other;   // global key row r+8 lives in half 1
    }

    // O^T += V^T @ P^T
#pragma unroll
    for (int kc = 0; kc < 4; ++kc) {
      const float bx = half ? brow[kc * 4 + 2] : brow[kc * 4 + 0];
      const float by = half ? brow[kc * 4 + 3] : brow[kc * 4 + 1];
      v2f bf = {bx, by};
      const int r0 = kc * 4 + 2 * half;
#pragma unroll
      for (int t = 0; t < 4; ++t) {
        v2f a = {sV[cur][r0 * 68 + 16 * t + col],
                 sV[cur][(r0 + 1) * 68 + 16 * t + col]};
        acc[t] = __builtin_amdgcn_wmma_f32_16x16x4_f32(
            false, a, false, bf, (short)0, acc[t], false, false);
      }
    }
    __syncthreads();   // all waves done with buf 'cur' before it is refilled
  }

  const float inv = 1.0f / l_run;
  const int s = q0 + col;
#pragma unroll
  for (int t = 0; t < 4; ++t)
#pragma unroll
    for (int r = 0; r < 8; ++r) {
      const int dd = 16 * t + r + 8 * half;
      attn[(size_t)(b * SEQ + s) * DMODEL + h * HDIM + dd] = acc[t][r] * inv;
    }
}

// ---------------------------------------------------------------------------
extern "C" void kernel_launch(void* const* d_in, const int* in_sizes, int n_in,
                              void* d_out, int out_size, void* d_ws, size_t ws_size,
                              hipStream_t stream) {
  const float* q   = (const float*)d_in[0];
  const float* k   = (const float*)d_in[1];
  const float* v   = (const float*)d_in[2];
  const int*   pad = (const int*)d_in[3];
  const float* w_q = (const float*)d_in[4];
  const float* b_q = (const float*)d_in[5];
  const float* w_k = (const float*)d_in[6];
  const float* b_k = (const float*)d_in[7];
  const float* w_v = (const float*)d_in[8];
  const float* b_v = (const float*)d_in[9];
  const float* w_o = (const float*)d_in[10];
  const float* b_o = (const float*)d_in[11];
  float* out = (float*)d_out;

  float* ws = (float*)d_ws;
  const size_t per = (size_t)BATCH * NHEAD * SEQ * HDIM;  // 6,291,456 floats
  float* qhB  = ws;
  float* khB  = ws + per;
  float* vhB  = ws + 2 * per;
  float* attn = ws + 3 * per;

  dim3 ggrid(MTOK / 64, DMODEL / 64);   // (128, 12)
  dim3 gblk(128);
  gemm_wmma<<<ggrid, gblk, 0, stream>>>(q, w_q, b_q, qhB, 1);
  gemm_wmma<<<ggrid, gblk, 0, stream>>>(k, w_k, b_k, khB, 1);
  gemm_wmma<<<ggrid, gblk, 0, stream>>>(v, w_v, b_v, vhB, 1);

  flash_attn<<<dim3(SEQ / 64, BATCH * NHEAD), 128, 0, stream>>>(
      qhB, khB, vhB, pad, attn);

  gemm_wmma<<<ggrid, gblk, 0, stream>>>(attn, w_o, b_o, out, 0);
}